// GCNDeconvolution_15977278341604
// MI455X (gfx1250) — compile-verified
//
#include <hip/hip_runtime.h>
#include <cstdint>
#include <cstddef>

// ---------------------------------------------------------------------------
// GCN forward on MI455X (gfx1250, wave32).
//  - GEMMs use V_WMMA_F32_16X16X4_F32 (CDNA5 fp32 matrix pipe, exact fp32).
//  - Edge aggregation: L2-resident gather + global_atomic_add_f32 scatter.
//  - Workspace layout: [dinv: 20000 f32][bufA: 20000x512 f32][bufB: 20000x512 f32]
//    total ~82 MB.
// ---------------------------------------------------------------------------

#define N_NODES 20000
#define D_FEAT  512
#define D_OUT   128

typedef float v2f __attribute__((ext_vector_type(2)));
typedef float v8f __attribute__((ext_vector_type(8)));

// ---------------- small utility kernels ----------------

__global__ void zero_f32_kernel(float* __restrict__ p, int n) {
    int i = blockIdx.x * blockDim.x + threadIdx.x;
    if (i < n) p[i] = 0.0f;
}

__global__ void deg_accum_kernel(const int* __restrict__ dst, float* __restrict__ deg, int n_edges) {
    int e = blockIdx.x * blockDim.x + threadIdx.x;
    if (e < n_edges) atomicAdd(&deg[dst[e]], 1.0f);
}

// dinv = rsqrt(deg_edges + 1)   (the +1 is the self-loop; deg >= 1 always)
__global__ void dinv_kernel(float* __restrict__ deg_io, int n) {
    int i = blockIdx.x * blockDim.x + threadIdx.x;
    if (i < n) deg_io[i] = rsqrtf(deg_io[i] + 1.0f);
}

// out[i,:] = dinv[i]^2 * h[i,:]   (self-loop contribution, also initializes out)
__global__ void agg_init_kernel(const float* __restrict__ h, const float* __restrict__ dinv,
                                float* __restrict__ out, int total) {
    int idx = blockIdx.x * blockDim.x + threadIdx.x;
    if (idx < total) {
        int node = idx >> 9;              // /512
        float di = dinv[node];
        out[idx] = di * di * h[idx];
    }
}

// x = relu(x + b[col])
__global__ void bias_relu_kernel(float* __restrict__ x, const float* __restrict__ b, int total) {
    int idx = blockIdx.x * blockDim.x + threadIdx.x;
    if (idx < total) {
        float v = x[idx] + b[idx & (D_FEAT - 1)];
        x[idx] = v > 0.0f ? v : 0.0f;
    }
}

// ---------------- edge scatter: one wave32 per edge ----------------
// out[dst,:] += dinv[src]*dinv[dst] * h[src,:]   (512 features, float4 loads)
__global__ __launch_bounds__(256)
void scatter_edges_kernel(const float* __restrict__ h, float* __restrict__ out,
                          const int* __restrict__ src, const int* __restrict__ dst,
                          const float* __restrict__ dinv, int n_edges) {
    int wave = (blockIdx.x * blockDim.x + threadIdx.x) >> 5;
    int lane = threadIdx.x & 31;
    if (wave >= n_edges) return;
    int s = src[wave];
    int d = dst[wave];
    float norm = dinv[s] * dinv[d];
    const float4* hs = (const float4*)(h + (size_t)s * D_FEAT);
    float* od = out + (size_t)d * D_FEAT;
    __builtin_prefetch(hs + lane, 0, 0);   // global_prefetch_b8
#pragma unroll
    for (int it = 0; it < 4; ++it) {
        int chunk = it * 32 + lane;        // float4 index within the 512-wide row
        float4 v = hs[chunk];
        int f = chunk * 4;
        atomicAdd(od + f + 0, norm * v.x);
        atomicAdd(od + f + 1, norm * v.y);
        atomicAdd(od + f + 2, norm * v.z);
        atomicAdd(od + f + 3, norm * v.w);
    }
}

// ---------------- fp32 WMMA GEMM: C[M,NOUT] = A[M,K] @ W[K,NOUT] ----------------
// block = 128 threads (4 waves); each wave computes a 16(M) x 32(N) strip of C
// (two 16x16 accumulators).  grid = (M/16, NOUT/128).  M must be multiple of 16
// (20000 = 1250*16) so EXEC stays all-1s for WMMA.
//
// V_WMMA_F32_16X16X4_F32 operand layout (ISA 7.12.2):
//   A (16x4):  lanes 0-15 -> M=lane, VGPR0/1 = K=k,k+1 ; lanes 16-31 -> K=k+2,k+3
//   B (4x16):  VGPR0: lanes 0-15 = row K=k  (N=lane), lanes 16-31 = row K=k+2
//              VGPR1: lanes 0-15 = row K=k+1,        lanes 16-31 = row K=k+3
//   C (16x16): VGPR r: lanes 0-15 -> M=r, N=lane ; lanes 16-31 -> M=r+8, N=lane-16
template <int NOUT, bool ADD_BIAS, bool RELU>
__global__ __launch_bounds__(128)
void gemm_wmma_f32_kernel(const float* __restrict__ A, const float* __restrict__ W,
                          const float* __restrict__ bias, float* __restrict__ C,
                          int M, int K) {
    const int lane  = threadIdx.x & 31;
    const int waveI = threadIdx.x >> 5;
    const int half  = lane >> 4;          // 0: lanes 0-15, 1: lanes 16-31
    const int l16   = lane & 15;

    const int m0 = blockIdx.x * 16;
    const int n0 = blockIdx.y * 128 + waveI * 32;

    const int mrow = m0 + l16;
    const float* arow = A + (size_t)mrow * K;

    v8f acc0 = {};
    v8f acc1 = {};

#pragma unroll 4
    for (int k = 0; k < K; k += 4) {
        const int kk = k + 2 * half;                       // this half-wave's K pair
        v2f afrag = *(const v2f*)(arow + kk);              // A[mrow, kk], A[mrow, kk+1]

        const float* wb = W + (size_t)kk * NOUT + n0 + l16;
        v2f b0 = { wb[0],  wb[NOUT] };                     // cols n0+l16
        v2f b1 = { wb[16], wb[NOUT + 16] };                // cols n0+16+l16

        acc0 = __builtin_amdgcn_wmma_f32_16x16x4_f32(false, afrag, false, b0,
                                                     (short)0, acc0, false, false);
        acc1 = __builtin_amdgcn_wmma_f32_16x16x4_f32(false, afrag, false, b1,
                                                     (short)0, acc1, false, false);
    }

    const int crow0 = m0 + 8 * half;
    const int c0 = n0 + l16;
    const int c1 = n0 + 16 + l16;
    float bias0 = 0.0f, bias1 = 0.0f;
    if (ADD_BIAS) { bias0 = bias[c0]; bias1 = bias[c1]; }

#pragma unroll
    for (int i = 0; i < 8; ++i) {
        int r = crow0 + i;
        float v0 = acc0[i];
        float v1 = acc1[i];
        if (ADD_BIAS) { v0 += bias0; v1 += bias1; }
        if (RELU) { v0 = v0 > 0.0f ? v0 : 0.0f; v1 = v1 > 0.0f ? v1 : 0.0f; }
        C[(size_t)r * NOUT + c0] = v0;
        C[(size_t)r * NOUT + c1] = v1;
    }
}

// ---------------- row softmax over 128 columns ----------------
__global__ __launch_bounds__(128)
void softmax_rows128_kernel(const float* __restrict__ logits, float* __restrict__ out) {
    __shared__ float sm[4];
    int row = blockIdx.x;
    int t   = threadIdx.x;
    float v = logits[(size_t)row * D_OUT + t];

    // max across 128 lanes: intra-wave32 then across 4 waves via LDS
    float m = v;
#pragma unroll
    for (int off = 16; off > 0; off >>= 1) m = fmaxf(m, __shfl_xor(m, off));
    if ((t & 31) == 0) sm[t >> 5] = m;
    __syncthreads();
    m = fmaxf(fmaxf(sm[0], sm[1]), fmaxf(sm[2], sm[3]));
    __syncthreads();

    float e = expf(v - m);
    float s = e;
#pragma unroll
    for (int off = 16; off > 0; off >>= 1) s += __shfl_xor(s, off);
    if ((t & 31) == 0) sm[t >> 5] = s;
    __syncthreads();
    s = sm[0] + sm[1] + sm[2] + sm[3];

    out[(size_t)row * D_OUT + t] = e / s;
}

// ---------------------------------------------------------------------------

extern "C" void kernel_launch(void* const* d_in, const int* in_sizes, int n_in,
                              void* d_out, int out_size, void* d_ws, size_t ws_size,
                              hipStream_t stream) {
    const float* x   = (const float*)d_in[0];
    const int*   ei  = (const int*)  d_in[1];
    const float* W1  = (const float*)d_in[2];
    const float* b1  = (const float*)d_in[3];
    const float* W2  = (const float*)d_in[4];
    const float* b2  = (const float*)d_in[5];
    const float* Wf1 = (const float*)d_in[6];
    const float* bf1 = (const float*)d_in[7];
    const float* Wf2 = (const float*)d_in[8];
    const float* bf2 = (const float*)d_in[9];
    float* out = (float*)d_out;

    const int E = in_sizes[1] / 2;
    const int* src = ei;        // edge_index[0]
    const int* dst = ei + E;    // edge_index[1]

    char*  ws   = (char*)d_ws;
    float* dinv = (float*)ws;                                   // N floats
    float* bufA = (float*)(ws + 81920);                         // N x 512
    float* bufB = bufA + (size_t)N_NODES * D_FEAT;              // N x 512

    const int NH = N_NODES * D_FEAT;
    const dim3 blk256(256);
    const dim3 gemmBlk(128);
    const dim3 gemmGrid512(N_NODES / 16, D_FEAT / 128);         // 1250 x 4
    const dim3 gemmGrid128(N_NODES / 16, D_OUT  / 128);         // 1250 x 1
    const dim3 edgeGrid((E * 32 + 255) / 256);                  // 1 wave32 per edge
    const dim3 elemGrid((NH + 255) / 256);

    // --- normalization: dinv = rsqrt(deg + 1) ---
    zero_f32_kernel<<<dim3((N_NODES + 255) / 256), blk256, 0, stream>>>(dinv, N_NODES);
    deg_accum_kernel<<<dim3((E + 255) / 256), blk256, 0, stream>>>(dst, dinv, E);
    dinv_kernel<<<dim3((N_NODES + 255) / 256), blk256, 0, stream>>>(dinv, N_NODES);

    // --- GCN layer 1: bufB = relu(Agg(x @ W1) + b1) ---
    gemm_wmma_f32_kernel<D_FEAT, false, false>
        <<<gemmGrid512, gemmBlk, 0, stream>>>(x, W1, nullptr, bufA, N_NODES, D_FEAT);
    agg_init_kernel<<<elemGrid, blk256, 0, stream>>>(bufA, dinv, bufB, NH);
    scatter_edges_kernel<<<edgeGrid, blk256, 0, stream>>>(bufA, bufB, src, dst, dinv, E);
    bias_relu_kernel<<<elemGrid, blk256, 0, stream>>>(bufB, b1, NH);

    // --- GCN layer 2: bufB = relu(Agg(bufB @ W2) + b2) ---
    gemm_wmma_f32_kernel<D_FEAT, false, false>
        <<<gemmGrid512, gemmBlk, 0, stream>>>(bufB, W2, nullptr, bufA, N_NODES, D_FEAT);
    agg_init_kernel<<<elemGrid, blk256, 0, stream>>>(bufA, dinv, bufB, NH);
    scatter_edges_kernel<<<edgeGrid, blk256, 0, stream>>>(bufA, bufB, src, dst, dinv, E);
    bias_relu_kernel<<<elemGrid, blk256, 0, stream>>>(bufB, b2, NH);

    // --- FC1 (fused bias + relu): bufA = relu(bufB @ Wf1 + bf1) ---
    gemm_wmma_f32_kernel<D_FEAT, true, true>
        <<<gemmGrid512, gemmBlk, 0, stream>>>(bufB, Wf1, bf1, bufA, N_NODES, D_FEAT);

    // --- FC2 (fused bias): bufB[:, :128] = bufA @ Wf2 + bf2 ---
    gemm_wmma_f32_kernel<D_OUT, true, false>
        <<<gemmGrid128, gemmBlk, 0, stream>>>(bufA, Wf2, bf2, bufB, N_NODES, D_FEAT);

    // --- row softmax -> d_out ---
    softmax_rows128_kernel<<<dim3(N_NODES), dim3(128), 0, stream>>>(bufB, out);
}